// Non_Max_Suppression_82703890252269
// MI455X (gfx1250) — compile-verified
//
#include <hip/hip_runtime.h>
#include <stdint.h>

// ---------------- problem constants (match reference) ----------------
#define BATCH   4
#define NBOX    4096
#define NW      (NBOX / 32)     // 128 bitmask words per row
#define KSEL    50              // N_MAX_OBJECTS
#define NITER   50
#define TILE    256
#define NTILES  (NBOX / TILE)   // 16
#define P_THR   0.1f
#define OV_THR  0.3f

// ---------------- gfx1250 feature probes ----------------
#if defined(__has_builtin)
#  if __has_builtin(__builtin_amdgcn_global_load_async_to_lds_b32) && \
      __has_builtin(__builtin_amdgcn_s_wait_asynccnt)
#    define USE_ASYNC_LDS 1
#  endif
#  if __has_builtin(__builtin_amdgcn_ballot_w32)
#    define BALLOT32(p) __builtin_amdgcn_ballot_w32(p)
#  endif
#endif
#ifndef BALLOT32
#  define BALLOT32(p) ((uint32_t)__ballot(p))
#endif

#if defined(USE_ASYNC_LDS)
typedef __attribute__((address_space(1))) int g_int;
typedef __attribute__((address_space(3))) int l_int;
__device__ __forceinline__ void async_copy_b32(const float* g, float* l) {
  // global -> LDS async DMA, one dword per lane; tracked by ASYNCcnt
  __builtin_amdgcn_global_load_async_to_lds_b32(
      (g_int*)g, (l_int*)l, 0, 0);
}
#endif

// =====================================================================
// Kernel A: build bit-packed cluster mask.
// One wave (32 lanes) per row i; j-tiles of box coords are double-buffered
// in LDS via async global->LDS copies. 32 overlap tests -> ballot -> 1 word.
// =====================================================================
__global__ __launch_bounds__(256)
void build_mask_kernel(const float* __restrict__ bx, const float* __restrict__ by,
                       const float* __restrict__ bw, const float* __restrict__ bh,
                       uint32_t* __restrict__ mask) {
  __shared__ float tb[2][4][TILE];   // [buf][bx,by,bw,bh][j in tile]
  const int wave = threadIdx.x >> 5;
  const int lane = threadIdx.x & 31;
  const int blocksPerBatch = NBOX / 8;                  // 512
  const int b = blockIdx.x / blocksPerBatch;
  const int i = (blockIdx.x % blocksPerBatch) * 8 + wave;
  const int base = b * NBOX;

  const float bxi = bx[base + i], byi = by[base + i];
  const float bwi = bw[base + i], bhi = bh[base + i];
  const float x1i = bxi - 0.5f * bwi, x3i = bxi + 0.5f * bwi;
  const float y1i = byi - 0.5f * bhi, y3i = byi + 0.5f * bhi;
  const float ai  = bwi * bhi;
  uint32_t* row = mask + (size_t)(base + i) * NW;

  const float* srcs[4] = {bx + base, by + base, bw + base, bh + base};

  auto load_tile = [&](int buf, int t) {
    // 8 waves cooperatively stage one tile: each wave copies 32 elems/array
    const int off = t * TILE + wave * 32 + lane;
    const int lo  = wave * 32 + lane;
#if defined(USE_ASYNC_LDS)
    async_copy_b32(srcs[0] + off, &tb[buf][0][lo]);
    async_copy_b32(srcs[1] + off, &tb[buf][1][lo]);
    async_copy_b32(srcs[2] + off, &tb[buf][2][lo]);
    async_copy_b32(srcs[3] + off, &tb[buf][3][lo]);
#else
    tb[buf][0][lo] = srcs[0][off];
    tb[buf][1][lo] = srcs[1][off];
    tb[buf][2][lo] = srcs[2][off];
    tb[buf][3][lo] = srcs[3][off];
#endif
  };
  auto wait_tile = [&]() {
#if defined(USE_ASYNC_LDS)
    __builtin_amdgcn_s_wait_asynccnt(0);
#endif
    __syncthreads();
  };

  int buf = 0;
  load_tile(0, 0);
  wait_tile();
  for (int t = 0; t < NTILES; ++t) {
    if (t + 1 < NTILES) load_tile(buf ^ 1, t + 1);   // prefetch next tile async
#pragma unroll
    for (int sub = 0; sub < TILE / 32; ++sub) {
      const int jj = sub * 32 + lane;
      const float bxj = tb[buf][0][jj], byj = tb[buf][1][jj];
      const float bwj = tb[buf][2][jj], bhj = tb[buf][3][jj];
      const float x1j = bxj - 0.5f * bwj, x3j = bxj + 0.5f * bwj;
      const float y1j = byj - 0.5f * bhj, y3j = byj + 0.5f * bhj;
      float iw = fminf(x3i, x3j) - fmaxf(x1i, x1j);
      float ih = fminf(y3i, y3j) - fmaxf(y1i, y1j);
      iw = fmaxf(iw, 0.0f);
      ih = fmaxf(ih, 0.0f);
      const float ov = (iw * ih) / fminf(ai, bwj * bhj);
      const uint32_t bits = BALLOT32(ov > OV_THR);
      if (lane == 0) row[t * (TILE / 32) + sub] = bits;
    }
    if (t + 1 < NTILES) { wait_tile(); buf ^= 1; }
  }
}

// On-the-fly row word (fallback when workspace too small for the mask).
__device__ __forceinline__ uint32_t calc_row_word(
    float x1i, float x3i, float y1i, float y3i, float ai,
    const float* __restrict__ bx, const float* __restrict__ by,
    const float* __restrict__ bw, const float* __restrict__ bh,
    int base, int w) {
  uint32_t bits = 0;
  const int j0 = w << 5;
#pragma unroll 4
  for (int k = 0; k < 32; ++k) {
    const int j = base + j0 + k;
    const float bwj = bw[j], bhj = bh[j];
    const float bxj = bx[j], byj = by[j];
    const float x1j = bxj - 0.5f * bwj, x3j = bxj + 0.5f * bwj;
    const float y1j = byj - 0.5f * bhj, y3j = byj + 0.5f * bhj;
    float iw = fminf(x3i, x3j) - fmaxf(x1i, x1j);
    float ih = fminf(y3i, y3j) - fmaxf(y1i, y1j);
    const float inter = fmaxf(iw, 0.0f) * fmaxf(ih, 0.0f);
    if (inter / fminf(ai, bwj * bhj) > OV_THR) bits |= (1u << k);
  }
  return bits;
}

// =====================================================================
// Kernel B: one workgroup per batch. 50 NMS iterations + stable top-50
// selection + gather, fully fused; all state in LDS (~26 KB).
// =====================================================================
template <bool HAS_MASK>
__global__ __launch_bounds__(1024)
void nms_iter_topk_kernel(const float* __restrict__ prob,
                          const float* __restrict__ bx, const float* __restrict__ by,
                          const float* __restrict__ bw, const float* __restrict__ bh,
                          const uint32_t* __restrict__ mask, float* __restrict__ out) {
  __shared__ float    sc[NBOX];           // scores, later p_masked
  __shared__ unsigned possible[NW], newly[NW], chosen[NW];
  __shared__ float    rV[1024];
  __shared__ int      rI[1024];
  __shared__ int      selIdx[KSEL];

  const int b = blockIdx.x, t = threadIdx.x;
  const int base = b * NBOX;

  for (int q = 0; q < 4; ++q) sc[t + q * 1024] = prob[base + t + q * 1024];
  if (t < NW) chosen[t] = 0u;
  __syncthreads();
  if (t < NW) {                            // possible0 = prob > P_THRESHOLD
    unsigned w = 0;
    for (int k = 0; k < 32; ++k) w |= (sc[t * 32 + k] > P_THR) ? (1u << k) : 0u;
    possible[t] = w;
  }
  __syncthreads();

  const uint32_t* mb = HAS_MASK ? (mask + (size_t)base * NW) : nullptr;

  for (int it = 0; it < NITER; ++it) {
    if (t < NW) newly[t] = 0u;
    __syncthreads();
    for (int q = 0; q < 4; ++q) {
      const int i = t + q * 1024;
      if ((possible[i >> 5] >> (i & 31)) & 1u) {
        const float si = sc[i];
        float x1i, x3i, y1i, y3i, ai;
        const uint32_t* row = nullptr;
        if (HAS_MASK) {
          row = mb + (size_t)i * NW;
        } else {
          const float bxi = bx[base + i], byi = by[base + i];
          const float bwi = bw[base + i], bhi = bh[base + i];
          x1i = bxi - 0.5f * bwi; x3i = bxi + 0.5f * bwi;
          y1i = byi - 0.5f * bhi; y3i = byi + 0.5f * bhi;
          ai  = bwi * bhi;
        }
        // i is chosen iff argmax_j over (cluster row ∩ possible) of score is i
        // (argmax = first max, so j<i with sj==si also disqualifies i)
        bool ismax = true;
        for (int w = 0; w < NW && ismax; ++w) {
          const unsigned pw = possible[w];
          if (!pw) continue;
          const uint32_t rw = HAS_MASK ? row[w]
              : calc_row_word(x1i, x3i, y1i, y3i, ai, bx, by, bw, bh, base, w);
          uint32_t m = rw & pw;
          while (m) {
            const int j = (w << 5) + __builtin_ctz(m);
            m &= m - 1u;
            const float sj = sc[j];
            if (sj > si || (sj == si && j < i)) { ismax = false; break; }
          }
        }
        if (ismax) {
          atomicOr(&chosen[i >> 5], 1u << (i & 31));
          // suppress this box's whole neighborhood (only live bits matter)
          for (int w = 0; w < NW; ++w) {
            const unsigned pw = possible[w];
            if (!pw) continue;
            const uint32_t rw = HAS_MASK ? row[w]
                : calc_row_word(x1i, x3i, y1i, y3i, ai, bx, by, bw, bh, base, w);
            const unsigned nb = rw & pw;
            if (nb) atomicOr(&newly[w], nb);
          }
        }
      }
    }
    __syncthreads();
    if (t < NW) possible[t] &= ~newly[t];
    __syncthreads();
  }

  // p_masked = prob * chosen, in place
  for (int q = 0; q < 4; ++q) {
    const int i = t + q * 1024;
    if (!((chosen[i >> 5] >> (i & 31)) & 1u)) sc[i] = 0.0f;
  }
  if (t < NW) newly[t] = 0u;               // reuse as "taken" mask
  __syncthreads();

  // stable top-50: descending value, ascending index on ties (matches lax.top_k)
  for (int k = 0; k < KSEL; ++k) {
    float bv = -1.0f; int bi = NBOX;
    for (int q = 0; q < 4; ++q) {
      const int i = t + q * 1024;
      if (!((newly[i >> 5] >> (i & 31)) & 1u)) {
        const float v = sc[i];
        if (v > bv || (v == bv && i < bi)) { bv = v; bi = i; }
      }
    }
    rV[t] = bv; rI[t] = bi;
    __syncthreads();
    for (int s = 512; s > 0; s >>= 1) {
      if (t < s) {
        if (rV[t + s] > rV[t] || (rV[t + s] == rV[t] && rI[t + s] < rI[t])) {
          rV[t] = rV[t + s]; rI[t] = rI[t + s];
        }
      }
      __syncthreads();
    }
    if (t == 0) {
      const int idx = rI[0];
      selIdx[k] = idx;
      newly[idx >> 5] |= (1u << (idx & 31));
    }
    __syncthreads();
  }

  // gather: out = [p_masked | bx | by | bw | bh], each [B, KSEL]
  if (t < KSEL) {
    const int idx = selIdx[t];
    const int o = b * KSEL + t;
    out[o]                    = sc[idx];
    out[1 * BATCH * KSEL + o] = bx[base + idx];
    out[2 * BATCH * KSEL + o] = by[base + idx];
    out[3 * BATCH * KSEL + o] = bw[base + idx];
    out[4 * BATCH * KSEL + o] = bh[base + idx];
  }
}

extern "C" void kernel_launch(void* const* d_in, const int* in_sizes, int n_in,
                              void* d_out, int out_size, void* d_ws, size_t ws_size,
                              hipStream_t stream) {
  const float* prob = (const float*)d_in[0];
  const float* bx   = (const float*)d_in[1];
  const float* by   = (const float*)d_in[2];
  const float* bw   = (const float*)d_in[3];
  const float* bh   = (const float*)d_in[4];
  float* out = (float*)d_out;

  const size_t mask_bytes = (size_t)BATCH * NBOX * NW * sizeof(uint32_t); // 8 MB
  if (ws_size >= mask_bytes) {
    uint32_t* mask = (uint32_t*)d_ws;
    build_mask_kernel<<<BATCH * (NBOX / 8), 256, 0, stream>>>(bx, by, bw, bh, mask);
    nms_iter_topk_kernel<true><<<BATCH, 1024, 0, stream>>>(prob, bx, by, bw, bh, mask, out);
  } else {
    nms_iter_topk_kernel<false><<<BATCH, 1024, 0, stream>>>(prob, bx, by, bw, bh, nullptr, out);
  }
  (void)in_sizes; (void)n_in; (void)out_size;
}